// FIRLayer_40261023433398
// MI455X (gfx1250) — compile-verified
//
#include <hip/hip_runtime.h>

// Complex FIR (11-tap, SAME) as WMMA f32 16x16x4 GEMMs.
// One 16x16 D tile = 128 consecutive complex outputs:
//   row m  -> segment starting at t0 + 8m (16 segments per tile)
//   col j  -> position p = j>>1 within segment, component o = j&1 (re/im)
// A row m = raw interleaved float stream x[t0+8m-5 .. t0+8m+12]  (K = 36, 9 steps of K=4)
// B[k][j] = banded shifted-filter matrix built from phi (complex multiply signs).

typedef __attribute__((ext_vector_type(2))) float v2f;
typedef __attribute__((ext_vector_type(8))) float v8f;

#define FILTER_DIM      11
#define NPTS            262144
#define TILES_PER_ROW   2048        // NPTS / 128 complex outputs per tile
#define TILES_PER_WAVE  4
#define WAVES_PER_BLOCK 8
#define KSTEPS          9           // K = 36 = 18 complex halo window * 2 floats

template<bool CHECK>
__device__ __forceinline__ void do_tile(const float* __restrict__ Xb,
                                        float* __restrict__ Ob,
                                        long row_c,   // complex index of this lane's window start (may be <0 / >=N on edges)
                                        int tr, int m, int h,
                                        const v2f Bf[KSTEPS]) {
    v8f acc = {};
    #pragma unroll
    for (int kk = 0; kk < KSTEPS; ++kk) {
        // This lane's A fragment: floats (4kk+2h, 4kk+2h+1) of row m
        // == complex sample (2kk + h) of the window -> one contiguous 8B load.
        long ci = row_c + 2 * kk + h;
        v2f a;
        if (CHECK) {
            if (ci >= 0 && ci < (long)NPTS) {
                a = *(const v2f*)(Xb + 2 * ci);
            } else {
                a.x = 0.0f; a.y = 0.0f;   // zero-pad halo at batch edges
            }
        } else {
            a = *(const v2f*)(Xb + 2 * ci);
        }
        // D = A*B + C  (8 args: neg_a, A, neg_b, B, c_mod, C, reuse_a, reuse_b)
        acc = __builtin_amdgcn_wmma_f32_16x16x4_f32(
                false, a, false, Bf[kk], (short)0, acc, false, false);
    }
    // Store: D VGPR v, lane(m,h) -> float index tr*256 + 16*v + 128*h + m.
    // Each half-wave writes 64 contiguous bytes per v -> fully coalesced.
    float* o = Ob + (long)tr * 256 + 128 * h + m;
    #pragma unroll
    for (int v = 0; v < 8; ++v) {
        o[16 * v] = acc[v];
    }
}

__global__ __launch_bounds__(256)
void fir_wmma_kernel(const float* __restrict__ X,
                     const float* __restrict__ phi,
                     float* __restrict__ out) {
    const int lane = threadIdx.x & 31;
    const int wave = threadIdx.x >> 5;
    const int m    = lane & 15;   // A row / B,D column index
    const int h    = lane >> 4;   // lane half (K offset selector)

    // ---- Build the 9 B fragments once per wave (constant across all tiles) ----
    // B layout (4x16, KxN): VGPR r, lane-half h holds K = 4kk + 2h + r, N = m.
    // k even pair = one complex sample s = 2kk+h, components c = 0 (.x) / 1 (.y).
    // Column j=m: p = m>>1 (position), o = m&1 (re/im output). Tap t = s - p.
    //   real out: +wr*xr - wi*xi     imag out: +wi*xr + wr*xi
    const int p = m >> 1;
    const int o = m & 1;
    v2f Bf[KSTEPS];
    #pragma unroll
    for (int kk = 0; kk < KSTEPS; ++kk) {
        int s = 2 * kk + h;
        int t = s - p;
        float wr = 0.0f, wi = 0.0f;
        if (t >= 0 && t < FILTER_DIM) {
            wr = phi[2 * t];        // phi[t][0][0]
            wi = phi[2 * t + 1];    // phi[t][1][0]
        }
        v2f bv;
        bv.x = o ? wi : wr;         // weight on xr
        bv.y = o ? wr : -wi;        // weight on xi
        Bf[kk] = bv;
    }

    // ---- Tile sweep: each wave handles 4 consecutive tiles in one batch row ----
    long chunk = (long)blockIdx.x * WAVES_PER_BLOCK + wave;
    long tile0 = chunk * TILES_PER_WAVE;            // 2048 % 4 == 0: never straddles batches
    int  b     = (int)(tile0 / TILES_PER_ROW);
    int  tr0   = (int)(tile0 % TILES_PER_ROW);
    const float* Xb = X   + (long)b * (2 * NPTS);
    float*       Ob = out + (long)b * (2 * NPTS);

    #pragma unroll
    for (int i = 0; i < TILES_PER_WAVE; ++i) {
        int tr = tr0 + i;
        long row_c = (long)tr * 128 + 8 * m - 5;    // window start (complex idx)
        // Speculative prefetch one tile ahead (global_prefetch_b8).
        __builtin_prefetch(Xb + 2 * (row_c + 128), 0, 1);
        if (tr == 0 || tr == TILES_PER_ROW - 1) {
            do_tile<true >(Xb, Ob, row_c, tr, m, h, Bf);   // zero-padded halo
        } else {
            do_tile<false>(Xb, Ob, row_c, tr, m, h, Bf);   // fast interior path
        }
    }
}

extern "C" void kernel_launch(void* const* d_in, const int* in_sizes, int n_in,
                              void* d_out, int out_size, void* d_ws, size_t ws_size,
                              hipStream_t stream) {
    const float* X   = (const float*)d_in[0];   // (128, 262144, 2) f32
    const float* phi = (const float*)d_in[1];   // (11, 2, 1) f32
    float* out = (float*)d_out;                 // (128, 262144, 2) f32

    // 128 batches * 2048 tiles = 262144 tiles; 8 waves/block * 4 tiles/wave = 32 tiles/block
    const int total_tiles = 128 * TILES_PER_ROW;
    dim3 grid(total_tiles / (WAVES_PER_BLOCK * TILES_PER_WAVE));  // 8192
    dim3 block(256);
    hipLaunchKernelGGL(fir_wmma_kernel, grid, block, 0, stream, X, phi, out);
}